// Model_52767968198948
// MI455X (gfx1250) — compile-verified
//
#include <hip/hip_runtime.h>
#include <math.h>

#define SEQ  4096
#define HID  1024
#define NCLS 256

typedef __attribute__((ext_vector_type(2))) float v2f;
typedef __attribute__((ext_vector_type(8))) float v8f;
typedef int v4i __attribute__((vector_size(16)));

#if __has_builtin(__builtin_amdgcn_global_load_async_to_lds_b128) && \
    __has_builtin(__builtin_amdgcn_s_wait_asynccnt)
#define HAVE_ASYNC 1
#else
#define HAVE_ASYNC 0
#endif

#define AS1 __attribute__((address_space(1)))
#define AS3 __attribute__((address_space(3)))

// ---------------------------------------------------------------------------
// Embedding gather: xs[t,:] = embed[tokens[t],:]
// ---------------------------------------------------------------------------
__global__ __launch_bounds__(256)
void gather_rows(const int* __restrict__ tok, const float* __restrict__ embed,
                 float* __restrict__ xs) {
  const int t   = blockIdx.x;
  const int row = tok[t];
  const float4* src = (const float4*)(embed + (size_t)row * HID);
  float4*       dst = (float4*)(xs + (size_t)t * HID);
  dst[threadIdx.x] = src[threadIdx.x];   // 256 threads * 16B = 4KB row
}

// ---------------------------------------------------------------------------
// C[M,N] = A[M,K] @ W[N,K]^T + (b0[N] + b1[N]),  fp32 WMMA 16x16x4.
// Block = 256 threads = 8 waves. Block tile 32(M) x 64(N); wave -> one 16x16.
// K is consumed in KC=32 chunks double-buffered in LDS; the next chunk is
// brought in with GLOBAL_LOAD_ASYNC_TO_LDS_B128 (ASYNCcnt) while the current
// chunk feeds 8 WMMAs per wave. Rows padded to 36 floats to spread banks.
// Frag layouts (f32 16x16x4):
//   A (16x4): lane l: M = l&15, K = 2*(l>>4) + v       (v = vgpr 0..1)
//   B (4x16): lane l: N = l&15, K = 2*(l>>4) + v
//   D (16x16): lane l: N = l&15, M = 8*(l>>4) + v      (v = 0..7)
// ---------------------------------------------------------------------------
#define KC   32
#define KPAD 36   // padded row stride in floats

__device__ __forceinline__ void wait_async_done() {
#if HAVE_ASYNC
  __builtin_amdgcn_s_wait_asynccnt(0);
#endif
}

__global__ __launch_bounds__(256)
void wmma_gemm_bias(const float* __restrict__ A, const float* __restrict__ W,
                    const float* __restrict__ b0, const float* __restrict__ b1,
                    float* __restrict__ C, int M, int N, int K) {
  __shared__ __align__(16) float Asm[2][32 * KPAD];
  __shared__ __align__(16) float Bsm[2][64 * KPAD];

  const int tid  = threadIdx.x;
  const int wave = tid >> 5;
  const int lane = tid & 31;
  const int half = lane >> 4;
  const int l16  = lane & 15;
  const int m0 = blockIdx.y * 32;
  const int n0 = blockIdx.x * 64;

  // staging indices: each thread copies 16B pieces (1x A row-chunk, 2x B)
  const int lrow = tid >> 3;        // 0..31
  const int lcol = (tid & 7) * 4;   // 0,4,...,28

  auto preload = [&](int kc, int pb) {
#if HAVE_ASYNC
    __builtin_amdgcn_global_load_async_to_lds_b128(
        (AS1 v4i*)(A + (size_t)(m0 + lrow) * K + kc + lcol),
        (AS3 v4i*)(&Asm[pb][lrow * KPAD + lcol]), 0, 0);
    #pragma unroll
    for (int rep = 0; rep < 2; ++rep)
      __builtin_amdgcn_global_load_async_to_lds_b128(
          (AS1 v4i*)(W + (size_t)(n0 + lrow + 32 * rep) * K + kc + lcol),
          (AS3 v4i*)(&Bsm[pb][(lrow + 32 * rep) * KPAD + lcol]), 0, 0);
#else
    *(float4*)&Asm[pb][lrow * KPAD + lcol] =
        *(const float4*)(A + (size_t)(m0 + lrow) * K + kc + lcol);
    #pragma unroll
    for (int rep = 0; rep < 2; ++rep)
      *(float4*)&Bsm[pb][(lrow + 32 * rep) * KPAD + lcol] =
          *(const float4*)(W + (size_t)(n0 + lrow + 32 * rep) * K + kc + lcol);
#endif
  };

  const int mrow = (wave >> 2) * 16 + l16;   // LDS row for A fragment
  const int nrow = (wave & 3) * 16 + l16;    // LDS row for B fragment

  v8f acc = {};
  int buf = 0;

  preload(0, 0);
  wait_async_done();
  __syncthreads();

  for (int kc = 0; kc < K; kc += KC) {
    if (kc + KC < K) preload(kc + KC, buf ^ 1);   // overlap copy of next chunk

#if __has_builtin(__builtin_amdgcn_wmma_f32_16x16x4_f32)
    #pragma unroll
    for (int kk = 0; kk < KC; kk += 4) {
      const float2 av = *(const float2*)&Asm[buf][mrow * KPAD + kk + 2 * half];
      const float2 bv = *(const float2*)&Bsm[buf][nrow * KPAD + kk + 2 * half];
      v2f a; a.x = av.x; a.y = av.y;
      v2f b; b.x = bv.x; b.y = bv.y;
      acc = __builtin_amdgcn_wmma_f32_16x16x4_f32(
          /*neg_a=*/false, a, /*neg_b=*/false, b,
          /*c_mod=*/(short)0, acc, /*reuse_a=*/false, /*reuse_b=*/false);
    }
#else
    // Host-pass / fallback path (same output mapping, plain FMAs).
    for (int v = 0; v < 8; ++v) {
      const int mr = (wave >> 2) * 16 + v + 8 * half;
      float s = acc[v];
      for (int kk = 0; kk < KC; ++kk)
        s += Asm[buf][mr * KPAD + kk] * Bsm[buf][nrow * KPAD + kk];
      acc[v] = s;
    }
#endif

    wait_async_done();
    __syncthreads();
    buf ^= 1;
  }

  const float bias = b0[n0 + (wave & 3) * 16 + l16] + b1[n0 + (wave & 3) * 16 + l16];
  #pragma unroll
  for (int v = 0; v < 8; ++v)
    C[(size_t)(m0 + (wave >> 2) * 16 + v + 8 * half) * N +
      (n0 + (wave & 3) * 16 + l16)] = acc[v] + bias;
}

// ---------------------------------------------------------------------------
// Persistent LSTM recurrence over SEQ steps, 32 workgroups cooperating.
// WG j owns hidden units [j*Hs, (j+1)*Hs); per step it computes their 4 gates
// (rows n, H+n, 2H+n, 3H+n of w_hh), streams w_hh from L2 with float4 loads,
// stages h in LDS, and publishes its new h slice to a global double buffer.
// Cross-WG step barrier: monotonic agent-scope atomic counter.
// ---------------------------------------------------------------------------
__device__ __forceinline__ float sigf(float x) { return 1.f / (1.f + __expf(-x)); }
__device__ __forceinline__ float tanh_fast(float x) {
  float e = __expf(2.f * x);
  return (e - 1.f) / (e + 1.f);
}

template <int H>
__global__ __launch_bounds__(256)
void lstm_rec(const float* __restrict__ pre,    // [SEQ, 4H] (bias folded in)
              const float* __restrict__ w_hh,   // [4H, H]
              float* __restrict__ hs_out,       // [SEQ, H]
              float* __restrict__ hbuf,         // [2, H], zero-initialized
              float* __restrict__ h_fin, float* __restrict__ c_fin,
              unsigned* __restrict__ cnt) {
  constexpr int NWG = 32, BLOCK = 256;
  constexpr int Hs = H / NWG;       // hidden units per WG
  constexpr int G  = 4 * Hs;        // gate values per WG
  constexpr int T  = BLOCK / G;     // threads per gate value
  constexpr int CH = H / T;         // per-thread K chunk

  __shared__ float h_sh[H];
  __shared__ float red[BLOCK];
  __shared__ float gate_sh[G];
  __shared__ float c_sh[Hs];

  const int tid = threadIdx.x;
  const int wg  = blockIdx.x;
  const int g   = tid / T;          // gate value handled by this thread group
  const int r   = tid % T;          // rank within group (split-K)
  const int gate_type = g / Hs;
  const int n_local   = g % Hs;
  const int row = gate_type * H + wg * Hs + n_local;   // w_hh / pre row
  const float4* wrow = (const float4*)(w_hh + (size_t)row * H + r * CH);

  if (tid < Hs) c_sh[tid] = 0.f;

  for (int t = 0; t < SEQ; ++t) {
    // stage h_{t-1} into LDS
    const float* hcur = hbuf + (size_t)(t & 1) * H;
    for (int i = tid; i < H; i += BLOCK) h_sh[i] = hcur[i];
    __syncthreads();

    // split-K partial dot: w_hh[row, r*CH : (r+1)*CH] . h
    float acc = 0.f;
    const float4* hv = (const float4*)(h_sh + r * CH);
    #pragma unroll 8
    for (int k = 0; k < CH / 4; ++k) {
      float4 w = wrow[k], h = hv[k];
      acc += w.x * h.x + w.y * h.y + w.z * h.z + w.w * h.w;
    }
    red[tid] = acc;
    __syncthreads();
    #pragma unroll
    for (int s = T / 2; s > 0; s >>= 1) {
      if (r < s) red[tid] += red[tid + s];
      __syncthreads();
    }
    if (r == 0) {
      gate_sh[g] = red[tid] + pre[(size_t)t * (4 * H) + row];
      // hide next step's activation fetch under the barrier wait
      if (t + 1 < SEQ)
        __builtin_prefetch(&pre[(size_t)(t + 1) * (4 * H) + row], 0, 1);
    }
    __syncthreads();

    // elementwise LSTM cell for this WG's hidden slice
    if (tid < Hs) {
      const int n = tid;
      const float i_ = sigf(gate_sh[0 * Hs + n]);
      const float f_ = sigf(gate_sh[1 * Hs + n]);
      const float g_ = tanh_fast(gate_sh[2 * Hs + n]);
      const float o_ = sigf(gate_sh[3 * Hs + n]);
      const float c  = f_ * c_sh[n] + i_ * g_;
      c_sh[n] = c;
      const float h = o_ * tanh_fast(c);
      hbuf[(size_t)((t + 1) & 1) * H + wg * Hs + n] = h;
      hs_out[(size_t)t * H + wg * Hs + n] = h;
      if (t == SEQ - 1) {
        h_fin[wg * Hs + n] = h;
        c_fin[wg * Hs + n] = c;
      }
    }

    // publish h slice, then grid-wide step barrier (monotonic counter)
    __threadfence();
    __syncthreads();
    if (tid == 0) {
      __hip_atomic_fetch_add(cnt, 1u, __ATOMIC_ACQ_REL, __HIP_MEMORY_SCOPE_AGENT);
      const unsigned target = (unsigned)(t + 1) * NWG;
      while (__hip_atomic_load(cnt, __ATOMIC_ACQUIRE, __HIP_MEMORY_SCOPE_AGENT) < target)
        __builtin_amdgcn_s_sleep(2);
    }
    __syncthreads();
    __builtin_amdgcn_fence(__ATOMIC_ACQUIRE, "agent");
  }
}

// ---------------------------------------------------------------------------
__global__ void init_state(float* hbuf, unsigned* cnt, int n) {
  const int i = blockIdx.x * blockDim.x + threadIdx.x;
  if (i < n)  hbuf[i] = 0.f;
  if (i == 0) *cnt = 0u;
}

// ---------------------------------------------------------------------------
extern "C" void kernel_launch(void* const* d_in, const int* in_sizes, int n_in,
                              void* d_out, int out_size, void* d_ws, size_t ws_size,
                              hipStream_t stream) {
  const int*   tokens = (const int*)  d_in[0];
  const float* embed  = (const float*)d_in[1];
  const float* w_ih0  = (const float*)d_in[2];
  const float* w_hh0  = (const float*)d_in[3];
  const float* b_ih0  = (const float*)d_in[4];
  const float* b_hh0  = (const float*)d_in[5];
  const float* w_ih1  = (const float*)d_in[6];
  const float* w_hh1  = (const float*)d_in[7];
  const float* b_ih1  = (const float*)d_in[8];
  const float* b_hh1  = (const float*)d_in[9];
  const float* w_ih2  = (const float*)d_in[10];
  const float* w_hh2  = (const float*)d_in[11];
  const float* b_ih2  = (const float*)d_in[12];
  const float* b_hh2  = (const float*)d_in[13];
  float* out = (float*)d_out;
  float* ws  = (float*)d_ws;

  // workspace layout (floats)
  float*    xs   = ws;                          // 4096*1024
  float*    pre  = ws + (size_t)4194304;        // 4096*4096 (reused 3x)
  float*    hs0  = ws + (size_t)20971520;       // 4096*1024
  float*    hs1  = ws + (size_t)25165824;       // 4096*1024
  float*    hbuf = ws + (size_t)29360128;       // 2*1024
  unsigned* cnt  = (unsigned*)(ws + (size_t)29362176);

  // d_out layout: hs2 [4096*256] | h_stack [2*1024] | c_stack [2*1024] | h2 [256] | c2 [256]
  float* out_hs2 = out;
  float* out_h   = out + 1048576;
  float* out_c   = out + 1050624;
  float* out_h2  = out + 1052672;
  float* out_c2  = out + 1052928;

  gather_rows<<<SEQ, 256, 0, stream>>>(tokens, embed, xs);

  // layer 0
  wmma_gemm_bias<<<dim3(4 * HID / 64, SEQ / 32), 256, 0, stream>>>(
      xs, w_ih0, b_ih0, b_hh0, pre, SEQ, 4 * HID, HID);
  init_state<<<(2 * HID + 255) / 256, 256, 0, stream>>>(hbuf, cnt, 2 * HID);
  lstm_rec<HID><<<32, 256, 0, stream>>>(pre, w_hh0, hs0, hbuf, out_h, out_c, cnt);

  // layer 1
  wmma_gemm_bias<<<dim3(4 * HID / 64, SEQ / 32), 256, 0, stream>>>(
      hs0, w_ih1, b_ih1, b_hh1, pre, SEQ, 4 * HID, HID);
  init_state<<<(2 * HID + 255) / 256, 256, 0, stream>>>(hbuf, cnt, 2 * HID);
  lstm_rec<HID><<<32, 256, 0, stream>>>(pre, w_hh1, hs1, hbuf, out_h + HID, out_c + HID, cnt);

  // layer 2 (hidden 256) — hs2 written straight into d_out
  wmma_gemm_bias<<<dim3(4 * NCLS / 64, SEQ / 32), 256, 0, stream>>>(
      hs1, w_ih2, b_ih2, b_hh2, pre, SEQ, 4 * NCLS, HID);
  init_state<<<(2 * NCLS + 255) / 256, 256, 0, stream>>>(hbuf, cnt, 2 * NCLS);
  lstm_rec<NCLS><<<32, 256, 0, stream>>>(pre, w_hh2, out_hs2, hbuf, out_h2, out_c2, cnt);
}